// MutilBlockExtractor_39779987095835
// MI455X (gfx1250) — compile-verified
//
#include <hip/hip_runtime.h>
#include <cstdint>

#define KK   3
#define HS   64
#define WS   64
#define HF   64
#define WF   64
#define DS_C 256
#define BATB 4     // BATCHSIZE
#define NUM  4     // bs / BATCHSIZE
#define HO   192
#define WO   192
#define CPT  8     // channels per thread

__global__ __launch_bounds__(192) void mbe_kernel(
    const float* __restrict__ src,
    const float* __restrict__ flow,
    const float* __restrict__ msk,
    float* __restrict__ out)
{
  // [n*3 + {fy,fx,m}][64] rows for this (bb, ysrc)
  __shared__ float s_fm[12 * 64];

  const int t     = threadIdx.x;       // 0..191
  const int yo    = blockIdx.x % HO;
  const int bb    = blockIdx.x / HO;   // 0..3
  const int cbase = blockIdx.y * CPT;
  const int ysrc  = yo / KK;

  // ---- CDNA5 async stage: flow/mask rows -> LDS (192 x b128 loads) ----
  {
    const int seg = t >> 4;            // 0..11  (n*3 + a)
    const int q   = t & 15;            // float4 index within a 64-float row
    const int n   = seg / 3;
    const int a   = seg - n * 3;
    const int b   = n * BATB + bb;
    uint32_t ldsByte = (uint32_t)(uintptr_t)(&s_fm[0]) + (uint32_t)(t * 16);
    if (a < 2) {
      uint32_t goff = (uint32_t)((((b * 2 + a) * HF + ysrc) * WF + q * 4) * 4);
      asm volatile("global_load_async_to_lds_b128 %0, %1, %2 offset:0"
                   :: "v"(ldsByte), "v"(goff), "s"(flow) : "memory");
    } else {
      uint32_t goff = (uint32_t)(((b * HF + ysrc) * WF + q * 4) * 4);
      asm volatile("global_load_async_to_lds_b128 %0, %1, %2 offset:0"
                   :: "v"(ldsByte), "v"(goff), "s"(msk) : "memory");
    }
    asm volatile("s_wait_asynccnt 0" ::: "memory");
  }
  __syncthreads();

  const int xo   = t;
  const int xsrc = xo / KK;
  const int yoi  = yo - ysrc * KK - 1;   // yo%3 - 1
  const int xoi  = xo - xsrc * KK - 1;   // xo%3 - 1

  float W[NUM][4];
  int   O[NUM][4];
  float M[NUM];
  int   PB[NUM];

  #pragma unroll
  for (int n = 0; n < NUM; ++n) {
    const int b = n * BATB + bb;
    const float fy = s_fm[(n * 3 + 0) * 64 + xsrc];
    const float fx = s_fm[(n * 3 + 1) * 64 + xsrc];
    M[n]           = s_fm[(n * 3 + 2) * 64 + xsrc];
    const float py = (float)(ysrc + yoi) + fy;
    const float px = (float)(xsrc + xoi) + fx;
    const int y0 = (int)floorf(py);
    const int x0 = (int)floorf(px);
    #pragma unroll
    for (int dy = 0; dy < 2; ++dy) {
      #pragma unroll
      for (int dx = 0; dx < 2; ++dx) {
        const int yi = y0 + dy, xi = x0 + dx;
        const float wy = 1.0f - fabsf(py - (float)yi);
        const float wx = 1.0f - fabsf(px - (float)xi);
        const bool valid = (yi >= 0) && (yi < HS) && (xi >= 0) && (xi < WS);
        const int yc = yi < 0 ? 0 : (yi > HS - 1 ? HS - 1 : yi);
        const int xc = xi < 0 ? 0 : (xi > WS - 1 ? WS - 1 : xi);
        W[n][dy * 2 + dx] = valid ? wy * wx : 0.0f;
        O[n][dy * 2 + dx] = yc * WS + xc;
      }
    }
    PB[n] = b * (DS_C * HS * WS);
  }

  #pragma unroll 4
  for (int cc = 0; cc < CPT; ++cc) {
    const int c    = cbase + cc;
    const int cOff = c * (HS * WS);
    float acc = 0.0f;
    #pragma unroll
    for (int n = 0; n < NUM; ++n) {
      const float* base = src + PB[n] + cOff;
      float s = W[n][0] * base[O[n][0]];
      s = s + W[n][1] * base[O[n][1]];
      s = s + W[n][2] * base[O[n][2]];
      s = s + W[n][3] * base[O[n][3]];
      acc = acc + M[n] * s;                 // same order as reference: (Σ4 corners)·m, Σn
    }
    float* op = out + (((size_t)(bb * DS_C + c) * HO + yo) * WO + xo);
    __builtin_nontemporal_store(acc, op);   // 151 MB write-once stream: NT hint
  }
}

extern "C" void kernel_launch(void* const* d_in, const int* in_sizes, int n_in,
                              void* d_out, int out_size, void* d_ws, size_t ws_size,
                              hipStream_t stream) {
  const float* src  = (const float*)d_in[0];   // (16,256,64,64) f32
  const float* flow = (const float*)d_in[1];   // (16,2,64,64)   f32
  const float* msk  = (const float*)d_in[2];   // (16,1,64,64)   f32
  float* out = (float*)d_out;                  // (4,256,192,192) f32

  dim3 grid(BATB * HO, DS_C / CPT);            // 768 x 32 blocks
  dim3 block(192, 1, 1);                       // 6 wave32, one output row per block
  mbe_kernel<<<grid, block, 0, stream>>>(src, flow, msk, out);

  (void)in_sizes; (void)n_in; (void)out_size; (void)d_ws; (void)ws_size;
}